// VectorQuantizer_35854386987349
// MI455X (gfx1250) — compile-verified
//
#include <hip/hip_runtime.h>

typedef __attribute__((ext_vector_type(2))) float v2f;
typedef __attribute__((ext_vector_type(8))) float v8f;

#define K_CODES 512
#define D_DIM   16
#define S_SP    131072            // T*H*W = 32*64*64
#define N_ROWS  262144            // B*S = 2*131072
#define CS      (D_DIM * S_SP)    // batch stride in x (C*S)

// ---------------------------------------------------------------------------
// Kernel 1: zero the one-hot encodings region (store-bandwidth bound, ~537MB).
// Handles arbitrary (mis)alignment of the region inside d_out.
// ---------------------------------------------------------------------------
__global__ __launch_bounds__(256) void vq_zero_enc(float* __restrict__ enc, size_t n) {
    size_t head = ((16 - ((size_t)enc & 15)) & 15) >> 2;   // floats until 16B aligned
    if (head > n) head = n;
    size_t n4 = (n - head) >> 2;
    size_t tail_start = head + (n4 << 2);

    size_t i = (size_t)blockIdx.x * blockDim.x + threadIdx.x;
    size_t stride = (size_t)gridDim.x * blockDim.x;

    if (i < head) enc[i] = 0.0f;
    float4* p = (float4*)(enc + head);
    float4 z = make_float4(0.f, 0.f, 0.f, 0.f);
    for (size_t j = i; j < n4; j += stride) p[j] = z;
    if (i < (n - tail_start)) enc[tail_start + i] = 0.0f;
}

// ---------------------------------------------------------------------------
// Kernel 2: WMMA distance GEMM + argmin. One wave handles 16 rows of xf.
// score(n) = 0.5*||E[n]||^2 - x.E[n]  (argmin-equivalent to full distance)
// ---------------------------------------------------------------------------
__global__ __launch_bounds__(256) void vq_argmin(const float* __restrict__ X,
                                                 const float* __restrict__ E,
                                                 float* __restrict__ enc,
                                                 int* __restrict__ idxOut) {
    __shared__ float Elds[K_CODES * D_DIM];   // 32 KB
    __shared__ float e2h[K_CODES];            // 2 KB

    int tid = threadIdx.x;
    // cooperative copy of codebook into LDS (coalesced b128)
    {
        const float4* src = (const float4*)E;
        float4* dst = (float4*)Elds;
        for (int i = tid; i < K_CODES * D_DIM / 4; i += 256) dst[i] = src[i];
    }
    __syncthreads();
    for (int k = tid; k < K_CODES; k += 256) {
        float s = 0.f;
        #pragma unroll
        for (int d = 0; d < D_DIM; ++d) { float v = Elds[k * D_DIM + d]; s += v * v; }
        e2h[k] = 0.5f * s;
    }
    __syncthreads();

    int lane = tid & 31;
    int wave = tid >> 5;
    int m = lane & 15;        // A-row / B-col within tile
    int h = lane >> 4;        // half-wave selector
    int rowBase = (blockIdx.x * 8 + wave) * 16;
    int row  = rowBase + m;
    int brow = row >> 17;              // / S_SP
    int sp   = row & (S_SP - 1);
    const float* xp = X + (size_t)brow * CS + sp;

    // A-matrix 16x4 fp32 layout: lanes0-15 K={0,1}, lanes16-31 K={2,3}
    v2f a[4];
    #pragma unroll
    for (int c = 0; c < 4; ++c) {
        int d0 = 4 * c + 2 * h;
        a[c].x = xp[(size_t)d0 * S_SP];
        a[c].y = xp[(size_t)(d0 + 1) * S_SP];
    }

    float best[8];
    int   bidx[8];
    #pragma unroll
    for (int j = 0; j < 8; ++j) { best[j] = 3.4e38f; bidx[j] = 0; }

    for (int nt = 0; nt < K_CODES / 16; ++nt) {
        int n = nt * 16 + m;
        const float* ep = &Elds[n * D_DIM];
        v2f b[4];
        #pragma unroll
        for (int c = 0; c < 4; ++c) {      // B 4x16: same per-lane K pattern as A
            int d0 = 4 * c + 2 * h;
            b[c].x = ep[d0];
            b[c].y = ep[d0 + 1];
        }
        v8f acc = {};
        #pragma unroll
        for (int c = 0; c < 4; ++c) {
            acc = __builtin_amdgcn_wmma_f32_16x16x4_f32(
                false, a[c], false, b[c], (short)0, acc, false, false);
        }
        float sb = e2h[n];
        #pragma unroll
        for (int j = 0; j < 8; ++j) {      // lane covers output rows j + 8*h
            float sc = sb - acc[j];
            if (sc < best[j]) { best[j] = sc; bidx[j] = n; }
        }
    }

    // min-reduce across each 16-lane half (xor masks 8..1 stay in-half)
    #pragma unroll
    for (int off = 8; off >= 1; off >>= 1) {
        #pragma unroll
        for (int j = 0; j < 8; ++j) {
            float ov = __shfl_xor(best[j], off, 32);
            int   oi = __shfl_xor(bidx[j], off, 32);
            if (ov < best[j] || (ov == best[j] && oi < bidx[j])) {
                best[j] = ov; bidx[j] = oi;
            }
        }
    }

    if (m == 0) {   // lane 0 -> rows rowBase+0..7, lane 16 -> rows rowBase+8..15
        #pragma unroll
        for (int j = 0; j < 8; ++j) {
            int r  = rowBase + 8 * h + j;
            int ki = bidx[j];
            idxOut[r] = ki;
            enc[(size_t)r * K_CODES + ki] = 1.0f;   // scatter the one-hot 1s
        }
    }
}

// ---------------------------------------------------------------------------
// Kernel 3: quantized gather (writes q_out in [B,C,T,H,W] layout) + loss
// partial sums. Deterministic block-tree reduction, no float atomics.
// ---------------------------------------------------------------------------
__global__ __launch_bounds__(256) void vq_quant_loss(const float* __restrict__ X,
                                                     const float* __restrict__ E,
                                                     const int* __restrict__ idx,
                                                     float* __restrict__ qout,
                                                     float* __restrict__ partial) {
    __shared__ float red[256];
    size_t base = (size_t)blockIdx.x * (256 * 4) + threadIdx.x;
    float acc = 0.f;
    #pragma unroll
    for (int u = 0; u < 4; ++u) {
        size_t o = base + (size_t)u * 256;          // o < 4,194,304 by grid sizing
        int sp = (int)(o & (S_SP - 1));
        int c  = (int)((o >> 17) & 15);
        int b  = (int)(o >> 21);
        int n  = b * S_SP + sp;
        int k  = idx[n];
        float q = E[k * D_DIM + c];
        float d = q - X[o];                          // same layout as output
        qout[o] = q;                                 // straight-through forward == quantized
        acc += d * d;
    }
    red[threadIdx.x] = acc;
    __syncthreads();
    for (int st = 128; st > 0; st >>= 1) {
        if (threadIdx.x < st) red[threadIdx.x] += red[threadIdx.x + st];
        __syncthreads();
    }
    if (threadIdx.x == 0) partial[blockIdx.x] = red[0];
}

__global__ __launch_bounds__(256) void vq_final(const float* __restrict__ partial,
                                                int np, float* __restrict__ loss) {
    __shared__ float red[256];
    float acc = 0.f;
    for (int i = threadIdx.x; i < np; i += 256) acc += partial[i];
    red[threadIdx.x] = acc;
    __syncthreads();
    for (int st = 128; st > 0; st >>= 1) {
        if (threadIdx.x < st) red[threadIdx.x] += red[threadIdx.x + st];
        __syncthreads();
    }
    if (threadIdx.x == 0)
        *loss = red[0] * (1.25f / (float)(N_ROWS * D_DIM));  // q_loss + 0.25*e_loss
}

// ---------------------------------------------------------------------------
extern "C" void kernel_launch(void* const* d_in, const int* in_sizes, int n_in,
                              void* d_out, int out_size, void* d_ws, size_t ws_size,
                              hipStream_t stream) {
    const float* X = (const float*)d_in[0];   // [2,16,32,64,64] fp32
    const float* E = (const float*)d_in[1];   // [512,16] fp32

    float* out  = (float*)d_out;
    float* loss = out;                                     // [1]
    float* qout = out + 1;                                 // [2,16,32,64,64] = N*D
    float* enc  = out + 1 + (size_t)N_ROWS * D_DIM;        // [N, 512]

    int*   idx     = (int*)d_ws;                                           // N ints
    float* partial = (float*)((char*)d_ws + (size_t)N_ROWS * sizeof(int)); // 4096 f32

    size_t encFloats = (size_t)N_ROWS * K_CODES;
    vq_zero_enc  <<<32768, 256, 0, stream>>>(enc, encFloats);
    vq_argmin    <<<N_ROWS / 128, 256, 0, stream>>>(X, E, enc, idx);
    vq_quant_loss<<<4096, 256, 0, stream>>>(X, E, idx, qout, partial);
    vq_final     <<<1, 256, 0, stream>>>(partial, 4096, loss);
}